// Bidirectional_Temporal_Self_Attention_14912126451855
// MI455X (gfx1250) — compile-verified
//
#include <hip/hip_runtime.h>
#include <hip/hip_bf16.h>

// Problem constants
#define BN 16
#define BC 128
#define BT 30
#define BH 64
#define BW 44
#define HW 2816           // H*W
#define THW 84480         // T*H*W
#define CTHW 10813440     // C*T*H*W
#define PSPLIT 8          // channel splits for the reduction
#define CH_PER 16         // channels per partial block (128/8)
#define HW4 704           // HW / 4 (float4 count per (n,c,t) slice)

typedef __attribute__((ext_vector_type(2))) float v2f;
typedef __attribute__((ext_vector_type(8))) float v8f;

// ---------------------------------------------------------------------------
// Kernel 1: partial sums for y[n,t] = sum over (c,h,w). One block handles 16
// channels of one (n,t); writes one partial float. Pure fp32 streaming.
// ---------------------------------------------------------------------------
__global__ void k_pool_partial(const float* __restrict__ x,
                               float* __restrict__ partial) {
  const int bid = blockIdx.x;          // n*T*PSPLIT + t*PSPLIT + p
  const int p   = bid & (PSPLIT - 1);
  const int nt  = bid >> 3;
  const int t   = nt % BT;
  const int n   = nt / BT;

  const float* base = x + (size_t)n * CTHW + (size_t)t * HW
                        + (size_t)p * CH_PER * THW;

  float acc = 0.0f;
  for (int c = 0; c < CH_PER; ++c) {
    const float4* src = (const float4*)(base + (size_t)c * THW);
    // prefetch next channel slice (speculative; safe on OOB)
    __builtin_prefetch((const void*)(base + (size_t)(c + 1) * THW +
                                     (size_t)threadIdx.x * 4), 0, 0);
    for (int i = threadIdx.x; i < HW4; i += 256) {
      float4 v = src[i];
      acc += (v.x + v.y) + (v.z + v.w);
    }
  }
  // wave reduce (wave32)
  for (int o = 16; o; o >>= 1) acc += __shfl_xor(acc, o, 32);
  __shared__ float red[8];
  const int wave = threadIdx.x >> 5, lane = threadIdx.x & 31;
  if (lane == 0) red[wave] = acc;
  __syncthreads();
  if (threadIdx.x == 0) {
    float s = 0.0f;
    for (int i = 0; i < 8; ++i) s += red[i];
    partial[bid] = s;
  }
}

// ---------------------------------------------------------------------------
// Kernel 2: one wave32 per batch n. Finish mean, 3 branches of conv1d +
// rank-1-score softmax + (v @ attn) via V_WMMA_F32_16X16X4_F32, sigmoid, sum.
// gate[n*T+t] = g1+g2+g3.
// ---------------------------------------------------------------------------
__global__ void k_attention(const float* __restrict__ partial,
                            const float* __restrict__ wq1, const float* __restrict__ wk1, const float* __restrict__ wv1,
                            const float* __restrict__ wq2, const float* __restrict__ wk2, const float* __restrict__ wv2,
                            const float* __restrict__ wq3, const float* __restrict__ wk3, const float* __restrict__ wv3,
                            float* __restrict__ gate) {
  const int n    = blockIdx.x;
  const int lane = threadIdx.x;            // 0..31, single wave per block

  __shared__ float ysh[40];                // halo-padded y (3 zeros each side)
  __shared__ float qsh[32];
  __shared__ float vsh[32];
  __shared__ float out_sh[32];
  __shared__ float attn_sh[32][33];        // padded to 32 rows for WMMA K

  // finish the mean
  float y = 0.0f;
  if (lane < BT) {
    const float* pp = partial + (n * BT + lane) * PSPLIT;
    for (int i = 0; i < PSPLIT; ++i) y += pp[i];
    y *= (1.0f / (float)(BC * BH * BW));
  }
  ysh[lane] = 0.0f;
  if (lane < 8) ysh[32 + lane] = 0.0f;
  __syncthreads();
  if (lane < BT) ysh[3 + lane] = y;
  __syncthreads();

  const float* WQ[3] = {wq1, wq2, wq3};
  const float* WK[3] = {wk1, wk2, wk3};
  const float* WV[3] = {wv1, wv2, wv3};

  const int hi2   = (lane >= 16) ? 2 : 0;  // K sub-offset for A/B halves
  const int col15 = lane & 15;

  float g = 0.0f;
  for (int b = 0; b < 3; ++b) {
    const int ksz = 3 + 2 * b;
    const int pad = b + 1;
    // SAME conv1d, zero padding handled by ysh halo
    float q = 0.0f, k = 0.0f, v = 0.0f;
    for (int j = 0; j < ksz; ++j) {
      float yy = ysh[3 + lane + j - pad];
      q = fmaf(yy, WQ[b][j], q);
      k = fmaf(yy, WK[b][j], k);
      v = fmaf(yy, WV[b][j], v);
    }
    if (lane >= BT) v = 0.0f;              // zero pad v for t>=30
    qsh[lane] = q;
    vsh[lane] = v;
    __syncthreads();

    // softmax rows: attn[i][j] = softmax_j(q[i]*k[j]); lane = j
    for (int i = 0; i < BT; ++i) {
      float r = (lane < BT) ? qsh[i] * k : -3.0e38f;
      float m = r;
      for (int o = 16; o; o >>= 1) m = fmaxf(m, __shfl_xor(m, o, 32));
      float e = (lane < BT) ? __expf(r - m) : 0.0f;
      float s = e;
      for (int o = 16; o; o >>= 1) s += __shfl_xor(s, o, 32);
      attn_sh[i][lane] = e / s;            // 0 for lanes >= 30
    }
    attn_sh[30][lane] = 0.0f;              // pad rows for K=30,31
    attn_sh[31][lane] = 0.0f;
    __syncthreads();

    // out[t] = sum_j v[j] * attn[j][t]  ==  (1x32) @ (32x32)
    // Replicate v into all 16 A-rows; 2 column tiles x 8 K-steps of 16x16x4.
    // A layout (16x4 f32): lanes 0-15 K={0,1}, lanes 16-31 K={2,3}.
    // B layout (4x16 f32): lanes 0-15 N=l K={0,1}, lanes 16-31 N=l-16 K={2,3}.
    for (int c = 0; c < 2; ++c) {
      v8f acc = {};
      for (int s4 = 0; s4 < 8; ++s4) {
        v2f A, B;
        A.x = vsh[4 * s4 + hi2];
        A.y = vsh[4 * s4 + hi2 + 1];
        B.x = attn_sh[4 * s4 + hi2][c * 16 + col15];
        B.y = attn_sh[4 * s4 + hi2 + 1][c * 16 + col15];
        acc = __builtin_amdgcn_wmma_f32_16x16x4_f32(
            /*neg_a=*/false, A, /*neg_b=*/false, B,
            /*c_mod=*/(short)0, acc, /*reuse_a=*/false, /*reuse_b=*/false);
      }
      // D row M=0 lives in acc[0] on lanes 0..15 (N = lane)
      if (lane < 16) out_sh[c * 16 + lane] = acc[0];
    }
    __syncthreads();

    float o = out_sh[lane];
    g += 1.0f / (1.0f + __expf(-o));       // sigmoid
    __syncthreads();                       // before next branch reuses LDS
  }

  if (lane < BT) gate[n * BT + lane] = g;
}

// ---------------------------------------------------------------------------
// Kernel 3: out = x * gate[n,t]. One block per (n,c,t) slice of 2816 floats.
// Bandwidth-critical: float4 stream, uniform scalar gate per block.
// ---------------------------------------------------------------------------
__global__ void k_scale(const float* __restrict__ x,
                        const float* __restrict__ gate,
                        float* __restrict__ out) {
  const int slice = blockIdx.x;            // (n*C + c)*T + t
  const int t = slice % BT;
  const int n = slice / (BC * BT);
  const float g = gate[n * BT + t];        // uniform -> scalar load

  const float4* src = (const float4*)(x + (size_t)slice * HW);
  float4*       dst = (float4*)(out + (size_t)slice * HW);
  for (int i = threadIdx.x; i < HW4; i += 256) {
    float4 v = src[i];
    v.x *= g; v.y *= g; v.z *= g; v.w *= g;
    dst[i] = v;
  }
}

// ---------------------------------------------------------------------------
extern "C" void kernel_launch(void* const* d_in, const int* in_sizes, int n_in,
                              void* d_out, int out_size, void* d_ws, size_t ws_size,
                              hipStream_t stream) {
  const float* x   = (const float*)d_in[0];
  const float* wq1 = (const float*)d_in[1];
  const float* wk1 = (const float*)d_in[2];
  const float* wv1 = (const float*)d_in[3];
  const float* wq2 = (const float*)d_in[4];
  const float* wk2 = (const float*)d_in[5];
  const float* wv2 = (const float*)d_in[6];
  const float* wq3 = (const float*)d_in[7];
  const float* wk3 = (const float*)d_in[8];
  const float* wv3 = (const float*)d_in[9];
  float* out = (float*)d_out;

  float* partial = (float*)d_ws;                 // N*T*PSPLIT = 3840 floats
  float* gate    = partial + 4096;               // N*T = 480 floats

  // 1) pooled partial sums (reads x once)
  k_pool_partial<<<BN * BT * PSPLIT, 256, 0, stream>>>(x, partial);

  // 2) tiny attention on [16,30] with WMMA f32 16x16x4
  k_attention<<<BN, 32, 0, stream>>>(partial,
                                     wq1, wk1, wv1,
                                     wq2, wk2, wv2,
                                     wq3, wk3, wv3,
                                     gate);

  // 3) broadcast-scale (reads x again, writes out)
  k_scale<<<BN * BC * BT, 256, 0, stream>>>(x, gate, out);
}